// ButterflyRotation_19705309954512
// MI455X (gfx1250) — compile-verified
//
#include <hip/hip_runtime.h>

typedef float v2f __attribute__((ext_vector_type(2)));
typedef float v8f __attribute__((ext_vector_type(8)));
typedef unsigned int v4u __attribute__((ext_vector_type(4)));
typedef int v4i __attribute__((ext_vector_type(4)));
typedef int v8i __attribute__((ext_vector_type(8)));

#define DIM     4096
#define BATCH   8192
#define HALF    (DIM / 2)     // angles per layer
#define NTILE   256           // 16x16 tiles per stage
#define NMAT    768           // 3 stages * 256 matrices
#define ROWS    16            // batch rows per workgroup (WMMA N dim)
#define THREADS 256           // 8 waves of 32
#define RS      4100          // padded LDS row stride in floats (4096 + 4)

// ---------------------------------------------------------------------------
// Kernel 1: fold each group of 4 butterfly layers into a dense 16x16 rotation
// matrix, emitted directly in V_WMMA_F32_16X16X4_F32 A-fragment layout:
//   frag[kc*64 + lane*2 + par] = M[lane&15][kc*4 + (lane>>4)*2 + par]
// One block of 16 threads per matrix; thread i owns column i of M.
// ---------------------------------------------------------------------------
__global__ void build_mats_kernel(const float* __restrict__ angles,
                                  float* __restrict__ mats) {
  const int m     = blockIdx.x;     // 0..767
  const int i     = threadIdx.x;    // column 0..15
  const int stage = m >> 8;
  const int grp   = m & 255;

  float col[16];
#pragma unroll
  for (int j = 0; j < 16; ++j) col[j] = (j == i) ? 1.0f : 0.0f;

#pragma unroll
  for (int sub = 0; sub < 4; ++sub) {
    const int t = 1 << sub;          // local stride
    const int l = stage * 4 + sub;   // global layer
#pragma unroll
    for (int p = 0; p < 16; ++p) {
      if (p & t) continue;           // p must be a "left" slot
      // global left index of this pair
      int il;
      if (stage == 0)      il = grp * 16 + p;
      else if (stage == 1) il = (grp >> 4) * 256 + p * 16 + (grp & 15);
      else                 il = p * 256 + grp;
      // reference pair-index: (il >> (l+1)) << l | (il & (2^l - 1))
      const int q    = il >> (l + 1);
      const int o    = il & ((1 << l) - 1);
      const int pidx = (q << l) | o;
      const float ang = angles[l * HALF + pidx];
      float s, c;
      __sincosf(ang, &s, &c);
      const float lo = col[p], ro = col[p + t];
      col[p]     =  c * lo + s * ro;   // new_left  =  xl*c + xr*s
      col[p + t] = -s * lo + c * ro;   // new_right = -xl*s + xr*c
    }
  }

  // scatter column i into the A-fragment layout
  float* out    = mats + m * 256;
  const int kc  = i >> 2;
  const int hh  = (i >> 1) & 1;
  const int par = i & 1;
#pragma unroll
  for (int j = 0; j < 16; ++j)
    out[kc * 64 + (hh * 16 + j) * 2 + par] = col[j];
}

// ---------------------------------------------------------------------------
// Tensor Data Mover: one-row (1 x 4096 fp32) 2D descriptor, global <-> LDS.
// Group layouts per CDNA5 ISA 8.3/8.4: g0 = {count, lds_addr, global_addr,
// type=2}; g1 = {mask=0, data_size=4B, tensor_dim0=4096, tensor_dim1=1,
// tile_dim0=4096, tile_dim1=1, dim0_stride=4096}; groups 2/3 zero (<=2D).
// ---------------------------------------------------------------------------
__device__ __forceinline__ void tdm_row_xfer(bool store, const float* gptr,
                                             unsigned lds_byte_off) {
  const unsigned long long ga = (unsigned long long)(const void*)gptr;
  v4u g0;
  g0[0] = 1u;                                               // count=1 (valid)
  g0[1] = lds_byte_off;                                     // lds_addr
  g0[2] = (unsigned)ga;                                     // global_addr[31:0]
  g0[3] = (unsigned)((ga >> 32) & 0x01FFFFFFu) | (2u << 30);// addr[56:32]|type=2
  v8i g1;
  g1[0] = (int)(2u << 16);                 // workgroup_mask=0, data_size=4B
  g1[1] = (int)((DIM & 0xFFFFu) << 16);    // tensor_dim0 lo16 @bits63:48
  g1[2] = (int)(((DIM >> 16) & 0xFFFFu) | (1u << 16)); // dim0 hi | dim1 lo (=1)
  g1[3] = (int)((DIM & 0xFFFFu) << 16);    // dim1 hi (=0) | tile_dim0=4096
  g1[4] = (int)1u;                         // tile_dim1=1, tile_dim2=0
  g1[5] = (int)DIM;                        // tensor_dim0_stride lo32
  g1[6] = 0;                               // stride hi | dim1_stride lo
  g1[7] = 0;                               // dim1_stride hi
  const v4i z = {0, 0, 0, 0};
#if defined(__clang_major__) && (__clang_major__ >= 23)
  const v8i z8 = {0, 0, 0, 0, 0, 0, 0, 0};
  if (store) __builtin_amdgcn_tensor_store_from_lds(g0, g1, z, z, z8, 0);
  else       __builtin_amdgcn_tensor_load_to_lds(g0, g1, z, z, z8, 0);
#else
  if (store) __builtin_amdgcn_tensor_store_from_lds(g0, g1, z, z, 0);
  else       __builtin_amdgcn_tensor_load_to_lds(g0, g1, z, z, 0);
#endif
}

// ---------------------------------------------------------------------------
// Kernel 2: one block = 16 batch rows x 4096 features resident in LDS.
// TDM streams rows in/out; three stages of 256 dense 16x16 rotations via
// v_wmma_f32_16x16x4_f32:  D(16 out-elems x 16 batch-rows) = M * X^T.
// ---------------------------------------------------------------------------
__global__ void __launch_bounds__(THREADS)
butterfly_wmma_kernel(const float* __restrict__ x,
                      const float* __restrict__ mats,
                      float* __restrict__ out) {
  extern __shared__ float lds[];
  const int tid  = threadIdx.x;
  const int lane = tid & 31;
  const int wave = tid >> 5;
  const int n = lane & 15;   // batch row within tile (WMMA N / lane dim)
  const int h = lane >> 4;   // lane half selects K / M split
  const long rowBase = (long)blockIdx.x * ROWS;

  // ---- TDM: each wave DMAs 2 rows (16 KB each) into padded LDS rows ----
#pragma unroll
  for (int k = 0; k < 2; ++k) {
    const int r = wave * 2 + k;
    tdm_row_xfer(false, x + (rowBase + r) * (long)DIM,
                 (unsigned)(r * RS * sizeof(float)));
  }
  __builtin_amdgcn_s_wait_tensorcnt(0);
  __syncthreads();

  // ---- three stages of batched 16x16x16 GEMMs ----
#pragma unroll
  for (int stage = 0; stage < 3; ++stage) {
    const float* mbase = mats + stage * NTILE * 256;
    for (int tile = wave; tile < NTILE; tile += THREADS / 32) {
      // feature column of element q of this tile's 16-vector
      int cbase, cstride;
      if (stage == 0)      { cbase = tile * 16;                       cstride = 1;   }
      else if (stage == 1) { cbase = (tile >> 4) * 256 + (tile & 15); cstride = 16;  }
      else                 { cbase = tile;                            cstride = 256; }

      const float* frag = mbase + tile * 256;
      v8f acc = {0.f, 0.f, 0.f, 0.f, 0.f, 0.f, 0.f, 0.f};
#pragma unroll
      for (int kc = 0; kc < 4; ++kc) {
        const v2f a = *reinterpret_cast<const v2f*>(frag + kc * 64 + lane * 2);
        const int k0 = kc * 4 + h * 2;
        v2f b;
        b.x = lds[n * RS + cbase + (k0 + 0) * cstride];
        b.y = lds[n * RS + cbase + (k0 + 1) * cstride];
        acc = __builtin_amdgcn_wmma_f32_16x16x4_f32(
            /*neg_a=*/false, a, /*neg_b=*/false, b,
            /*c_mod=*/(short)0, acc, /*reuse_a=*/false, /*reuse_b=*/false);
      }
      // D: lane -> batch row n; VGPR i -> output element j = i + 8*h
#pragma unroll
      for (int i = 0; i < 8; ++i) {
        const int j = i + 8 * h;
        lds[n * RS + cbase + j * cstride] = acc[i];
      }
    }
    __syncthreads();   // next stage reads columns written by all waves
  }

  // ---- TDM: stream result rows back out ----
#pragma unroll
  for (int k = 0; k < 2; ++k) {
    const int r = wave * 2 + k;
    tdm_row_xfer(true, out + (rowBase + r) * (long)DIM,
                 (unsigned)(r * RS * sizeof(float)));
  }
  __builtin_amdgcn_s_wait_tensorcnt(0);
}

// ---------------------------------------------------------------------------
extern "C" void kernel_launch(void* const* d_in, const int* in_sizes, int n_in,
                              void* d_out, int out_size, void* d_ws, size_t ws_size,
                              hipStream_t stream) {
  const float* x      = (const float*)d_in[0];
  const float* angles = (const float*)d_in[1];
  // d_in[2]/d_in[3] (left_idx/right_idx) are implied by the analytic butterfly
  // structure and are not needed.
  float* mats = (float*)d_ws;        // 768 * 256 floats = 768 KB scratch
  float* out  = (float*)d_out;

  build_mats_kernel<<<NMAT, 16, 0, stream>>>(angles, mats);
  butterfly_wmma_kernel<<<BATCH / ROWS, THREADS, ROWS * RS * sizeof(float), stream>>>(
      x, mats, out);
}